// VehicleModeler_42210938585813
// MI455X (gfx1250) — compile-verified
//
#include <hip/hip_runtime.h>

#define B_TOTAL 8192
#define T_STEPS 256

typedef float v2f __attribute__((ext_vector_type(2)));
typedef float v8f __attribute__((ext_vector_type(8)));

// D = A(16x4) * B(4x16) + C, fp32 WMMA (CDNA5 V_WMMA_F32_16X16X4_F32)
static __device__ __forceinline__ v8f wmma4(v2f a, v2f b, v8f c) {
  return __builtin_amdgcn_wmma_f32_16x16x4_f32(false, a, false, b, (short)0, c,
                                               false, false);
}

// full 16x16x16 matmul as a K-chain of 4 WMMAs
static __device__ __forceinline__ v8f mm16(const v2f a[4], const v2f b[4], v8f acc) {
#pragma unroll
  for (int k = 0; k < 4; ++k) acc = wmma4(a[k], b[k], acc);
  return acc;
}

// Swap the two 16-lane halves of the wave: pure VALU (v_permlanex16_b32),
// avoids ds_bpermute + s_wait_dscnt on the serial recurrent chain.
static __device__ __forceinline__ float xor16(float v) {
#if __has_builtin(__builtin_amdgcn_permlanex16)
  unsigned u = __builtin_bit_cast(unsigned, v);
  unsigned r = __builtin_amdgcn_permlanex16(u, u, 0x76543210u, 0xFEDCBA98u,
                                            false, false);
  return __builtin_bit_cast(float, r);
#else
  return __shfl_xor(v, 16, 32);
#endif
}

// Hardware transcendentals (CDNA5 V_TANH_F32 / V_RCP_F32 / V_RSQ_F32)
static __device__ __forceinline__ float fast_tanh(float x) {
#if __has_builtin(__builtin_amdgcn_tanhf)
  return __builtin_amdgcn_tanhf(x);
#else
  return tanhf(x);
#endif
}
static __device__ __forceinline__ float fast_rcp(float x) {
#if __has_builtin(__builtin_amdgcn_rcpf)
  return __builtin_amdgcn_rcpf(x);
#else
  return 1.0f / x;
#endif
}
static __device__ __forceinline__ float fast_rsq(float x) {
#if __has_builtin(__builtin_amdgcn_rsqf)
  return __builtin_amdgcn_rsqf(x);
#else
  return rsqrtf(x);
#endif
}
static __device__ __forceinline__ float sigf(float x) {
  return fast_rcp(1.0f + __expf(-x));
}

// Convert a matrix held in f32 C/D layout (reg r = row r (lanes<16) / r+8
// (lanes>=16), col = lane%16) into four B-operand chunks (rows 4k..4k+3).
static __device__ __forceinline__ void cd_to_b(const float x[8], bool lo, v2f b[4]) {
  float p2 = xor16(x[2]), p3 = xor16(x[3]);
  b[0][0] = lo ? x[0] : p2;   // rows 0 / 2
  b[0][1] = lo ? x[1] : p3;   // rows 1 / 3
  float p6 = xor16(x[6]), p7 = xor16(x[7]);
  b[1][0] = lo ? x[4] : p6;   // rows 4 / 6
  b[1][1] = lo ? x[5] : p7;   // rows 5 / 7
  float p0 = xor16(x[0]), p1 = xor16(x[1]);
  b[2][0] = lo ? p0 : x[2];   // rows 8 / 10
  b[2][1] = lo ? p1 : x[3];   // rows 9 / 11
  float p4 = xor16(x[4]), p5 = xor16(x[5]);
  b[3][0] = lo ? p4 : x[6];   // rows 12 / 14
  b[3][1] = lo ? p5 : x[7];   // rows 13 / 15
}

__global__ __launch_bounds__(128)
void vehicle_lstm_wmma(const float* __restrict__ initial_state,
                       const float* __restrict__ commands,
                       const float* __restrict__ ln_gamma,
                       const float* __restrict__ ln_beta,
                       const float* __restrict__ W_in,
                       const float* __restrict__ b_in,
                       const float* __restrict__ W_ih,
                       const float* __restrict__ W_hh,
                       const float* __restrict__ b_ih,
                       const float* __restrict__ b_hh,
                       const float* __restrict__ W_out,
                       const float* __restrict__ b_out,
                       float* __restrict__ out) {
  const int lane = threadIdx.x & 31;
  const int wave = threadIdx.x >> 5;
  const bool lo  = lane < 16;
  const int  n   = lane & 15;                       // sample column + A-row
  const int  b   = blockIdx.x * 64 + wave * 16 + n; // batch element
  const int  kof = lo ? 0 : 2;                      // A/B K sub-offset per half
  const int  rb  = lo ? 0 : 8;                      // C/D row base per half

  // ---- preload weights into A-matrix layout (constant for whole kernel) ----
  v2f a_in[4], a_ih[4][4], a_hh[4][4], a_out[4];
#pragma unroll
  for (int k = 0; k < 4; ++k) {
    const int kb = 4 * k + kof;
    a_in[k][0] = W_in[n * 16 + kb];
    a_in[k][1] = W_in[n * 16 + kb + 1];
#pragma unroll
    for (int g = 0; g < 4; ++g) {
      a_ih[g][k][0] = W_ih[(g * 16 + n) * 16 + kb];
      a_ih[g][k][1] = W_ih[(g * 16 + n) * 16 + kb + 1];
      a_hh[g][k][0] = W_hh[(g * 16 + n) * 16 + kb];
      a_hh[g][k][1] = W_hh[(g * 16 + n) * 16 + kb + 1];
    }
    a_out[k][0] = (n < 12) ? W_out[n * 16 + kb] : 0.0f;     // pad rows 12-15
    a_out[k][1] = (n < 12) ? W_out[n * 16 + kb + 1] : 0.0f;
  }

  // ---- biases / LN params in C/D row layout ----
  v8f bias_in, bias_g[4], bias_out;
  float gam[8], bet[8];
#pragma unroll
  for (int r = 0; r < 8; ++r) {
    const int d = r + rb;
    gam[r] = ln_gamma[d];
    bet[r] = ln_beta[d];
    bias_in[r] = b_in[d];
#pragma unroll
    for (int g = 0; g < 4; ++g)
      bias_g[g][r] = b_ih[g * 16 + d] + b_hh[g * 16 + d];
    bias_out[r] = (d < 12) ? b_out[d] : 0.0f;
  }

  // ---- initial state (rows 0-7 in lo lanes, 8-11 in hi lanes) ----
  float s[8];
  const float* st0 = initial_state + (size_t)b * 12;
#pragma unroll
  for (int r = 0; r < 8; ++r) {
    float vlo = st0[r];
    float vhi = (r < 4) ? st0[8 + r] : 0.0f;
    s[r] = lo ? vlo : vhi;
  }

  float cst[8];                 // LSTM cell state, C/D layout
  v2f bh[4];                    // h in B-operand layout (starts at 0)
#pragma unroll
  for (int r = 0; r < 8; ++r) cst[r] = 0.0f;
#pragma unroll
  for (int k = 0; k < 4; ++k) { bh[k][0] = 0.0f; bh[k][1] = 0.0f; }

  const float4* cmd4 = reinterpret_cast<const float4*>(commands);

  for (int t = 0; t < T_STEPS; ++t) {
    // x = concat(state, cmd): rows 0-11 state, rows 12-15 command
    float4 c4 = cmd4[(size_t)b * T_STEPS + t];  // lanes n and n+16 share a line
    float x[8];
    x[0] = s[0]; x[1] = s[1]; x[2] = s[2]; x[3] = s[3];
    x[4] = lo ? s[4] : c4.x;
    x[5] = lo ? s[5] : c4.y;
    x[6] = lo ? s[6] : c4.z;
    x[7] = lo ? s[7] : c4.w;

    // LayerNorm over the 16 rows of each column
    float sum = 0.0f, sq = 0.0f;
#pragma unroll
    for (int r = 0; r < 8; ++r) { sum += x[r]; sq += x[r] * x[r]; }
    sum += xor16(sum);
    sq  += xor16(sq);
    const float mu   = sum * 0.0625f;
    const float var  = sq * 0.0625f - mu * mu;
    const float rstd = fast_rsq(var + 1e-5f);
#pragma unroll
    for (int r = 0; r < 8; ++r) x[r] = (x[r] - mu) * rstd * gam[r] + bet[r];

    // x1 = W_in @ x + b_in
    v2f bmat[4];
    cd_to_b(x, lo, bmat);
    v8f x1 = mm16(a_in, bmat, bias_in);

    float x1a[8];
#pragma unroll
    for (int r = 0; r < 8; ++r) x1a[r] = x1[r];
    cd_to_b(x1a, lo, bmat);

    // gates = W_ih @ x1 + W_hh @ h + (b_ih + b_hh)
    v8f Gi = mm16(a_hh[0], bh, mm16(a_ih[0], bmat, bias_g[0]));
    v8f Gf = mm16(a_hh[1], bh, mm16(a_ih[1], bmat, bias_g[1]));
    v8f Gg = mm16(a_hh[2], bh, mm16(a_ih[2], bmat, bias_g[2]));
    v8f Go = mm16(a_hh[3], bh, mm16(a_ih[3], bmat, bias_g[3]));

    float hnew[8];
#pragma unroll
    for (int r = 0; r < 8; ++r) {
      float cc = sigf(Gf[r]) * cst[r] + sigf(Gi[r]) * fast_tanh(Gg[r]);
      cst[r]   = cc;
      hnew[r]  = sigf(Go[r]) * fast_tanh(cc);
    }

    // state += W_out @ h + b_out  (rows 0-11 only)
    cd_to_b(hnew, lo, bh);  // bh reused next step for W_hh @ h
    v8f ds = mm16(a_out, bh, bias_out);
#pragma unroll
    for (int r = 0; r < 8; ++r) {
      float ns = s[r] + ds[r];
      s[r] = (lo || r < 4) ? ns : s[r];
    }

    // normalize sin/cos pairs (rows 1&2, 3&4, 5&6) — lo-half rows only
#pragma unroll
    for (int p = 0; p < 3; ++p) {
      const int ci = 1 + 2 * p, si = 2 + 2 * p;
      float cs = s[ci], sn = s[si];
      float inv = fast_rcp(sqrtf(sn * sn + cs * cs) + 1e-8f);
      s[ci] = lo ? cs * inv : s[ci];
      s[si] = lo ? sn * inv : s[si];
    }

    // emit state for this timestep: 12 floats, 16B-aligned float4 stores
    float* op = out + ((size_t)b * T_STEPS + t) * 12;
    if (lo) {
      *reinterpret_cast<float4*>(op)     = make_float4(s[0], s[1], s[2], s[3]);
      *reinterpret_cast<float4*>(op + 4) = make_float4(s[4], s[5], s[6], s[7]);
    } else {
      *reinterpret_cast<float4*>(op + 8) = make_float4(s[0], s[1], s[2], s[3]);
    }
  }
}

extern "C" void kernel_launch(void* const* d_in, const int* in_sizes, int n_in,
                              void* d_out, int out_size, void* d_ws, size_t ws_size,
                              hipStream_t stream) {
  (void)in_sizes; (void)n_in; (void)out_size; (void)d_ws; (void)ws_size;
  dim3 grid(B_TOTAL / 64);   // 4 waves/block * 16 samples/wave = 64 samples
  dim3 block(128);
  vehicle_lstm_wmma<<<grid, block, 0, stream>>>(
      (const float*)d_in[0],  // initial_state
      (const float*)d_in[1],  // commands
      (const float*)d_in[2],  // ln_gamma
      (const float*)d_in[3],  // ln_beta
      (const float*)d_in[4],  // W_in
      (const float*)d_in[5],  // b_in
      (const float*)d_in[6],  // W_ih
      (const float*)d_in[7],  // W_hh
      (const float*)d_in[8],  // b_ih
      (const float*)d_in[9],  // b_hh
      (const float*)d_in[10], // W_out
      (const float*)d_in[11], // b_out
      (float*)d_out);
}